// GATConv_343597384438
// MI455X (gfx1250) — compile-verified
//
#include <hip/hip_runtime.h>
#include <hip/hip_bf16.h>

typedef float v2f __attribute__((ext_vector_type(2)));
typedef float v4f __attribute__((ext_vector_type(4)));
typedef float v8f __attribute__((ext_vector_type(8)));

#define F_IN      256
#define D_OUT     64
#define NEG_SLOPE 0.2f

// ---- monotone float<->uint encoding so segment-max can use u32 atomics ----
__device__ __forceinline__ unsigned fenc(float f) {
    unsigned u = __float_as_uint(f);
    return (u & 0x80000000u) ? ~u : (u | 0x80000000u);
}
__device__ __forceinline__ float fdec(unsigned u) {
    return (u & 0x80000000u) ? __uint_as_float(u & 0x7FFFFFFFu)
                             : __uint_as_float(~u);
}
#define FENC_NEG_INF 0x007FFFFFu  // fenc(-inf)

// ---------------------------------------------------------------------------
// Kernel 0: zero the output accumulator (b128 stores), init node max/sum
// ---------------------------------------------------------------------------
__global__ void gat_init(v4f* __restrict__ out4, unsigned* __restrict__ nmax,
                         float* __restrict__ nsum, size_t oelems4, int n_nodes) {
    const size_t stride = (size_t)gridDim.x * blockDim.x;
    const v4f z = {};
    for (size_t i = (size_t)blockIdx.x * blockDim.x + threadIdx.x; i < oelems4; i += stride)
        out4[i] = z;
    for (size_t i = (size_t)blockIdx.x * blockDim.x + threadIdx.x; i < (size_t)n_nodes; i += stride) {
        nmax[i] = FENC_NEG_INF;
        nsum[i] = 0.0f;
    }
}

// ---------------------------------------------------------------------------
// Kernel 1: h = feat @ W^T  via V_WMMA_F32_16X16X4_F32
//   one wave per 32-row node tile (2 WMMA M-tiles) -> B fragments reused 2x.
//   A 16x4 f32 fragment: lane(l) row M=l&15, K = {0,1}+2*(l>>4)  -> v2f load
//   B  4x16 f32 fragment (B = W^T): mirrored layout, b = W[n*256 + kk..kk+1]
//   C/D: VGPR r of lane l holds (M = r + 8*(l>>4), N = l&15)
//   Per k-step: 2 A loads + 4 B loads (all b64) feed 8 v_wmma.
// ---------------------------------------------------------------------------
__global__ __launch_bounds__(256) void gat_gemm_wmma(
        const float* __restrict__ feat, const float* __restrict__ W,
        float* __restrict__ h, int n_nodes, int n_tiles32) {
    const int wave = blockIdx.x * (blockDim.x >> 5) + (threadIdx.x >> 5);
    if (wave >= n_tiles32) return;

    const int lane = threadIdx.x & 31;
    const int half = lane >> 4;     // 0: K pair {0,1}, 1: K pair {2,3}
    const int lrow = lane & 15;
    const int row_base = wave * 32;

    int ar0 = row_base + lrow;
    int ar1 = row_base + 16 + lrow;
    if (ar0 >= n_nodes) ar0 = n_nodes - 1;     // clamp (duplicate reads OK)
    if (ar1 >= n_nodes) ar1 = n_nodes - 1;
    const float* arow0 = feat + (size_t)ar0 * F_IN + half * 2;
    const float* arow1 = feat + (size_t)ar1 * F_IN + half * 2;
    const float* brow  = W + (size_t)lrow * F_IN + half * 2;

    v8f acc0[4] = {};    // M-tile 0 (rows row_base .. +15), N-tiles 0..3
    v8f acc1[4] = {};    // M-tile 1 (rows row_base+16 .. +31)

    for (int k = 0; k < F_IN; k += 4) {
        const v2f a0 = *(const v2f*)(arow0 + k);
        const v2f a1 = *(const v2f*)(arow1 + k);
#pragma unroll
        for (int t = 0; t < 4; ++t) {
            const v2f b = *(const v2f*)(brow + (size_t)t * 16 * F_IN + k);
            acc0[t] = __builtin_amdgcn_wmma_f32_16x16x4_f32(
                false, a0, false, b, (short)0, acc0[t], false, false);
            acc1[t] = __builtin_amdgcn_wmma_f32_16x16x4_f32(
                false, a1, false, b, (short)0, acc1[t], false, false);
        }
    }

    // store D: row = row_base + mt*16 + half*8 + r, col = t*16 + lrow
    if (row_base + 32 <= n_nodes) {
        // fast path: whole 32-row tile in range, no per-element predication
        float* o0 = h + (size_t)(row_base + half * 8) * D_OUT + lrow;
        float* o1 = o0 + (size_t)16 * D_OUT;
#pragma unroll
        for (int t = 0; t < 4; ++t) {
#pragma unroll
            for (int r = 0; r < 8; ++r) {
                o0[(size_t)r * D_OUT + t * 16] = acc0[t][r];
                o1[(size_t)r * D_OUT + t * 16] = acc1[t][r];
            }
        }
    } else {
        const int m0 = row_base + half * 8;
#pragma unroll
        for (int t = 0; t < 4; ++t) {
#pragma unroll
            for (int r = 0; r < 8; ++r) {
                if (m0 + r < n_nodes)
                    h[(size_t)(m0 + r) * D_OUT + t * 16 + lrow] = acc0[t][r];
                if (m0 + 16 + r < n_nodes)
                    h[(size_t)(m0 + 16 + r) * D_OUT + t * 16 + lrow] = acc1[t][r];
            }
        }
    }
}

// ---------------------------------------------------------------------------
// Kernel 2: per-node attention scores el/er (wave per node, shuffle reduce)
// ---------------------------------------------------------------------------
__global__ __launch_bounds__(256) void gat_node_scores(
        const float* __restrict__ h, const float* __restrict__ al,
        const float* __restrict__ ar, float* __restrict__ el,
        float* __restrict__ er, int n_nodes) {
    const int node = blockIdx.x * (blockDim.x >> 5) + (threadIdx.x >> 5);
    if (node >= n_nodes) return;
    const int lane = threadIdx.x & 31;

    const v2f hv = *(const v2f*)(h + (size_t)node * D_OUT + 2 * lane);
    const v2f lv = *(const v2f*)(al + 2 * lane);
    const v2f rv = *(const v2f*)(ar + 2 * lane);
    float sl = hv[0] * lv[0] + hv[1] * lv[1];
    float sr = hv[0] * rv[0] + hv[1] * rv[1];
#pragma unroll
    for (int off = 16; off > 0; off >>= 1) {
        sl += __shfl_xor(sl, off, 32);
        sr += __shfl_xor(sr, off, 32);
    }
    if (lane == 0) {
        el[node] = sl;
        er[node] = sr;
    }
}

// ---------------------------------------------------------------------------
// Kernel 3: edge logits + segment-max over dst (u32 atomic, monotone coding)
// ---------------------------------------------------------------------------
__global__ void gat_edge_logits(const int* __restrict__ src, const int* __restrict__ dst,
                                const float* __restrict__ el, const float* __restrict__ er,
                                float* __restrict__ ebuf, unsigned* __restrict__ nmax,
                                int n_edges) {
    const int i = blockIdx.x * blockDim.x + threadIdx.x;
    if (i >= n_edges) return;
    const float x = el[src[i]] + er[dst[i]];
    const float e = (x > 0.0f) ? x : NEG_SLOPE * x;
    ebuf[i] = e;
    atomicMax(&nmax[dst[i]], fenc(e));
}

// ---------------------------------------------------------------------------
// Kernel 4: ee = exp(e - max[dst]); segment-sum over dst
// ---------------------------------------------------------------------------
__global__ void gat_edge_exp(const int* __restrict__ dst, const unsigned* __restrict__ nmax,
                             float* __restrict__ ebuf, float* __restrict__ nsum,
                             int n_edges) {
    const int i = blockIdx.x * blockDim.x + threadIdx.x;
    if (i >= n_edges) return;
    const int v = dst[i];
    const float t = __expf(ebuf[i] - fdec(nmax[v]));
    ebuf[i] = t;
    atomicAdd(&nsum[v], t);
}

// ---------------------------------------------------------------------------
// Kernel 5: rst[dst] += (ee/sum[dst]) * h[src]
//   wave per edge; lane owns an adjacent pair -> h gathered as coalesced b64
// ---------------------------------------------------------------------------
__global__ __launch_bounds__(256) void gat_aggregate(
        const int* __restrict__ src, const int* __restrict__ dst,
        const float* __restrict__ ebuf, const float* __restrict__ nsum,
        const float* __restrict__ h, float* __restrict__ out, int n_edges) {
    const int e = blockIdx.x * (blockDim.x >> 5) + (threadIdx.x >> 5);
    if (e >= n_edges) return;
    const int lane = threadIdx.x & 31;
    const int u = src[e];
    const int v = dst[e];
    const float a = ebuf[e] / nsum[v];
    const v2f hv = *(const v2f*)(h + (size_t)u * D_OUT + 2 * lane);
    float* orow = out + (size_t)v * D_OUT + 2 * lane;
    atomicAdd(&orow[0], a * hv[0]);
    atomicAdd(&orow[1], a * hv[1]);
}

// ---------------------------------------------------------------------------
extern "C" void kernel_launch(void* const* d_in, const int* in_sizes, int n_in,
                              void* d_out, int out_size, void* d_ws, size_t ws_size,
                              hipStream_t stream) {
    const float* feat = (const float*)d_in[0];
    const float* W    = (const float*)d_in[1];
    const float* al   = (const float*)d_in[2];
    const float* ar   = (const float*)d_in[3];
    const int*   src  = (const int*)d_in[4];
    const int*   dst  = (const int*)d_in[5];
    float* out = (float*)d_out;

    const int N = in_sizes[0] / F_IN;   // 100000
    const int E = in_sizes[4];          // 1600000

    // workspace carve (all 256B aligned): h, el, er, nmax, nsum, ebuf (~33.6MB)
    char* ws = (char*)d_ws;
    size_t off = 0;
    auto carve = [&](size_t bytes) -> void* {
        void* p = ws + off;
        off = (off + bytes + 255) & ~(size_t)255;
        return p;
    };
    float*    h    = (float*)carve((size_t)N * D_OUT * sizeof(float));
    float*    el   = (float*)carve((size_t)N * sizeof(float));
    float*    er   = (float*)carve((size_t)N * sizeof(float));
    unsigned* nmax = (unsigned*)carve((size_t)N * sizeof(unsigned));
    float*    nsum = (float*)carve((size_t)N * sizeof(float));
    float*    ebuf = (float*)carve((size_t)E * sizeof(float));
    (void)ws_size;

    const size_t oelems4 = (size_t)N * D_OUT / 4;
    gat_init<<<2048, 256, 0, stream>>>((v4f*)out, nmax, nsum, oelems4, N);

    const int tiles32 = (N + 31) / 32;
    gat_gemm_wmma<<<(tiles32 + 7) / 8, 256, 0, stream>>>(feat, W, h, N, tiles32);

    gat_node_scores<<<(N + 7) / 8, 256, 0, stream>>>(h, al, ar, el, er, N);

    gat_edge_logits<<<(E + 255) / 256, 256, 0, stream>>>(src, dst, el, er, ebuf, nmax, E);
    gat_edge_exp<<<(E + 255) / 256, 256, 0, stream>>>(dst, nmax, ebuf, nsum, E);
    gat_aggregate<<<(E + 7) / 8, 256, 0, stream>>>(src, dst, ebuf, nsum, h, out, E);
}